// GraphConvolution_953482740189
// MI455X (gfx1250) — compile-verified
//
#include <hip/hip_runtime.h>
#include <hip/hip_bf16.h>

typedef __attribute__((ext_vector_type(2))) float v2f;
typedef __attribute__((ext_vector_type(8))) float v8f;
typedef __attribute__((ext_vector_type(4))) float f4;

#define DIM 256

// ---------------------------------------------------------------------------
// Kernel 1: support = X @ W using fp32 WMMA (V_WMMA_F32_16X16X4_F32).
// One block = one 16-row tile of X; 8 waves, wave w computes cols [32w,32w+32).
// A-frag (16x4 f32): lane L<16 -> M=L, K={k0,k0+1}; lane>=16 -> K={k0+2,k0+3}.
// B-frag (4x16): lane n<16 -> N=n, K={k0,k0+1}; lanes 16-31 -> K={k0+2,k0+3}.
// C/D: VGPR r -> M = r + 8*(lane>=16), N = lane&15.
// ---------------------------------------------------------------------------
__global__ __launch_bounds__(256) void gcn_gemm(const float* __restrict__ X,
                                                const float* __restrict__ W,
                                                float* __restrict__ S)
{
    const int lane = threadIdx.x & 31;
    const int wave = threadIdx.x >> 5;
    const int half = lane >> 4;
    const int l16  = lane & 15;
    const int row0 = blockIdx.x << 4;
    const int c0   = wave << 5;

    const float* Arow = X + (size_t)(row0 + l16) * DIM;

    v8f acc0 = {};
    v8f acc1 = {};

    for (int k0 = 0; k0 < DIM; k0 += 4) {
        const int ka = k0 + (half << 1);
        v2f a = *(const v2f*)(Arow + ka);

        const float* Wk = W + (size_t)ka * DIM;
        v2f b0, b1;
        b0.x = Wk[c0 + l16];        b0.y = Wk[DIM + c0 + l16];
        b1.x = Wk[c0 + 16 + l16];   b1.y = Wk[DIM + c0 + 16 + l16];

        acc0 = __builtin_amdgcn_wmma_f32_16x16x4_f32(false, a, false, b0,
                                                     (short)0, acc0, false, false);
        acc1 = __builtin_amdgcn_wmma_f32_16x16x4_f32(false, a, false, b1,
                                                     (short)0, acc1, false, false);
    }

#pragma unroll
    for (int r = 0; r < 8; ++r) {
        const int m = r + (half << 3);
        float* dst = S + (size_t)(row0 + m) * DIM + c0 + l16;
        dst[0]  = acc0[r];
        dst[16] = acc1[r];
    }
}

// ------------------------- CSR construction ---------------------------------

__global__ __launch_bounds__(256) void k_zero(int* __restrict__ cnt, int n)
{
    int i = blockIdx.x * blockDim.x + threadIdx.x;
    if (i < n) cnt[i] = 0;
}

__global__ __launch_bounds__(256) void k_count(const int* __restrict__ rowi,
                                               int* __restrict__ cnt, int nEdges)
{
    int e = blockIdx.x * blockDim.x + threadIdx.x;
    if (e >= nEdges) return;
    (void)__hip_atomic_fetch_add(&cnt[rowi[e]], 1,
                                 __ATOMIC_RELAXED, __HIP_MEMORY_SCOPE_AGENT);
}

// Scan pass 1: per-block (1024 elements) exclusive scan + block totals.
__global__ __launch_bounds__(256) void k_scan1(const int* __restrict__ cnt,
                                               int* __restrict__ off,
                                               int* __restrict__ bsum, int n)
{
    __shared__ int sh[256];
    const int t = threadIdx.x;
    const int base = blockIdx.x * 1024 + t * 4;
    int v[4];
    int s = 0;
#pragma unroll
    for (int j = 0; j < 4; ++j) {
        int idx = base + j;
        v[j] = (idx < n) ? cnt[idx] : 0;
        s += v[j];
    }
    sh[t] = s;
    __syncthreads();
    for (int d = 1; d < 256; d <<= 1) {
        int x = (t >= d) ? sh[t - d] : 0;
        __syncthreads();
        sh[t] += x;
        __syncthreads();
    }
    if (t == 255) bsum[blockIdx.x] = sh[255];
    int run = sh[t] - s;                       // exclusive prefix of this chunk
#pragma unroll
    for (int j = 0; j < 4; ++j) {
        int idx = base + j;
        if (idx < n) off[idx] = run;
        run += v[j];
    }
}

// Scan pass 2: exclusive scan of block totals (nb <= 1024), one block.
__global__ __launch_bounds__(1024) void k_scan2(int* __restrict__ bsum, int nb)
{
    __shared__ int sh[1024];
    const int t = threadIdx.x;
    int v = (t < nb) ? bsum[t] : 0;
    sh[t] = v;
    __syncthreads();
    for (int d = 1; d < 1024; d <<= 1) {
        int x = (t >= d) ? sh[t - d] : 0;
        __syncthreads();
        sh[t] += x;
        __syncthreads();
    }
    if (t < nb) bsum[t] = sh[t] - v;           // exclusive
}

// Scan pass 3: add block offsets; also copy into write pointers.
__global__ __launch_bounds__(256) void k_scan3(int* __restrict__ off,
                                               const int* __restrict__ bsum,
                                               int* __restrict__ wp,
                                               int n, int nEdges)
{
    int i = blockIdx.x * blockDim.x + threadIdx.x;
    if (i == 0) off[n] = nEdges;
    if (i >= n) return;
    int o = off[i] + bsum[i >> 10];
    off[i] = o;
    wp[i]  = o;
}

// Permute (col, val) into row-grouped order.
__global__ __launch_bounds__(256) void k_scatter(const int* __restrict__ rowi,
                                                 const int* __restrict__ coli,
                                                 const float* __restrict__ vals,
                                                 int* __restrict__ wp,
                                                 int* __restrict__ colp,
                                                 float* __restrict__ valp,
                                                 int nEdges)
{
    int e = blockIdx.x * blockDim.x + threadIdx.x;
    if (e >= nEdges) return;
    int p = __hip_atomic_fetch_add(&wp[rowi[e]], 1,
                                   __ATOMIC_RELAXED, __HIP_MEMORY_SCOPE_AGENT);
    colp[p] = coli[e];
    valp[p] = vals[e];
}

// ---------------------------------------------------------------------------
// Kernel: CSR SpMM. One wave per output row. Accumulates in 8 VGPRs/lane
// (initialized from bias, so the bias add is fused), writes each output row
// once with plain stores — zero f32 atomics. Prefetches the next edge's
// source row (global_prefetch_b8) to hide L2 latency.
// ---------------------------------------------------------------------------
__global__ __launch_bounds__(256) void gcn_spmm_csr(const float* __restrict__ S,
                                                    const int* __restrict__ off,
                                                    const int* __restrict__ colp,
                                                    const float* __restrict__ valp,
                                                    const float* __restrict__ bias,
                                                    float* __restrict__ out,
                                                    int nNodes)
{
    const int lane = threadIdx.x & 31;
    const int r = blockIdx.x * (blockDim.x >> 5) + (threadIdx.x >> 5);
    if (r >= nNodes) return;

    const int jb = off[r];
    const int je = off[r + 1];
    const int o0 = lane << 2;
    const int o1 = 128 + (lane << 2);

    f4 a0 = *(const f4*)(bias + o0);
    f4 a1 = *(const f4*)(bias + o1);

    for (int j = jb; j < je; ++j) {
        const int   c = colp[j];
        const float v = valp[j];
        const float* src = S + (size_t)c * DIM;
        if (j + 1 < je) {
            const float* nxt = S + (size_t)colp[j + 1] * DIM;
            __builtin_prefetch(nxt + o0, 0, 3);
        }
        f4 m0 = *(const f4*)(src + o0);
        f4 m1 = *(const f4*)(src + o1);
#pragma unroll
        for (int q = 0; q < 4; ++q) {
            a0[q] = fmaf(v, m0[q], a0[q]);
            a1[q] = fmaf(v, m1[q], a1[q]);
        }
    }

    float* dst = out + (size_t)r * DIM;
    *(f4*)(dst + o0) = a0;
    *(f4*)(dst + o1) = a1;
}

// --------------------- Fallback (atomic) path kernels -----------------------

__global__ __launch_bounds__(256) void gcn_bias_init(float* __restrict__ out,
                                                     const float* __restrict__ bias,
                                                     int total4)
{
    int i = blockIdx.x * blockDim.x + threadIdx.x;
    if (i >= total4) return;
    int c4 = i & (DIM / 4 - 1);
    f4 b = *(const f4*)(bias + (c4 << 2));
    *(f4*)(out + ((size_t)i << 2)) = b;
}

__global__ __launch_bounds__(256) void gcn_spmm_atomic(const float* __restrict__ S,
                                                       const float* __restrict__ vals,
                                                       const int* __restrict__ rowi,
                                                       const int* __restrict__ coli,
                                                       float* __restrict__ out,
                                                       int nEdges)
{
    const int lane = threadIdx.x & 31;
    const int e = blockIdx.x * (blockDim.x >> 5) + (threadIdx.x >> 5);
    if (e >= nEdges) return;

    const int   r = rowi[e];
    const int   c = coli[e];
    const float v = vals[e];

    const float* src = S + (size_t)c * DIM;
    float*       dst = out + (size_t)r * DIM;

    f4 m0 = *(const f4*)(src + (lane << 2));
    f4 m1 = *(const f4*)(src + 128 + (lane << 2));

#pragma unroll
    for (int j = 0; j < 4; ++j) {
        (void)__hip_atomic_fetch_add(dst + (lane << 2) + j, v * m0[j],
                                     __ATOMIC_RELAXED, __HIP_MEMORY_SCOPE_AGENT);
        (void)__hip_atomic_fetch_add(dst + 128 + (lane << 2) + j, v * m1[j],
                                     __ATOMIC_RELAXED, __HIP_MEMORY_SCOPE_AGENT);
    }
}

// ---------------------------------------------------------------------------
// inputs: 0=x [N,256] f32, 1=adj_vals [E] f32, 2=row [E] i32, 3=col [E] i32,
//         4=weight [256,256] f32, 5=bias [256] f32.   out: [N,256] f32.
// ws layout: S [N*256 f32] | cnt [N] | off [N+1] | wp [N] | colp [E] |
//            valp [E] | bsum [1024]
// ---------------------------------------------------------------------------
extern "C" void kernel_launch(void* const* d_in, const int* in_sizes, int n_in,
                              void* d_out, int out_size, void* d_ws, size_t ws_size,
                              hipStream_t stream) {
    const float* x    = (const float*)d_in[0];
    const float* adjv = (const float*)d_in[1];
    const int*   row  = (const int*)d_in[2];
    const int*   col  = (const int*)d_in[3];
    const float* w    = (const float*)d_in[4];
    const float* bias = (const float*)d_in[5];
    float*       out  = (float*)d_out;

    const int nNodes = in_sizes[0] / DIM;   // 100000 (divisible by 16)
    const int nEdges = in_sizes[1];         // 3200000

    float* S    = (float*)d_ws;
    int*   cnt  = (int*)(S + (size_t)nNodes * DIM);
    int*   off  = cnt + nNodes;
    int*   wp   = off + nNodes + 1;
    int*   colp = wp + nNodes;
    float* valp = (float*)(colp + nEdges);
    int*   bsum = (int*)(valp + nEdges);

    const size_t needCSR = (size_t)nNodes * DIM * 4           // S
                         + (size_t)(3 * nNodes + 1) * 4       // cnt/off/wp
                         + (size_t)nEdges * 8                 // colp/valp
                         + 4096;                              // bsum

    // 1) support = X @ W  (fp32 WMMA)
    gcn_gemm<<<nNodes / 16, 256, 0, stream>>>(x, w, S);

    if (ws_size >= needCSR) {
        // 2) CSR build: histogram -> exclusive scan -> permute edges
        k_zero<<<(nNodes + 255) / 256, 256, 0, stream>>>(cnt, nNodes);
        k_count<<<(nEdges + 255) / 256, 256, 0, stream>>>(row, cnt, nEdges);

        const int nb = (nNodes + 1023) / 1024;        // 98 blocks
        k_scan1<<<nb, 256, 0, stream>>>(cnt, off, bsum, nNodes);
        k_scan2<<<1, 1024, 0, stream>>>(bsum, nb);
        k_scan3<<<(nNodes + 255) / 256, 256, 0, stream>>>(off, bsum, wp,
                                                          nNodes, nEdges);
        k_scatter<<<(nEdges + 255) / 256, 256, 0, stream>>>(row, col, adjv,
                                                            wp, colp, valp, nEdges);

        // 3) SpMM: wave per row, register accumulation, bias fused, no atomics
        gcn_spmm_csr<<<(nNodes + 7) / 8, 256, 0, stream>>>(S, off, colp, valp,
                                                           bias, out, nNodes);
    } else {
        // Fallback: atomic scatter-add formulation
        const int total4 = nNodes * (DIM / 4);
        gcn_bias_init<<<(total4 + 255) / 256, 256, 0, stream>>>(out, bias, total4);
        gcn_spmm_atomic<<<(nEdges + 7) / 8, 256, 0, stream>>>(S, adjv, row, col,
                                                              out, nEdges);
    }
}